// Radon_3358664426199
// MI455X (gfx1250) — compile-verified
//
#include <hip/hip_runtime.h>
#include <math.h>
#include <stdint.h>

#define RES      256
#define A_TOT    256
#define BATCH    8
#define A_BLK    8            // angles per workgroup
#define STRIDE   260          // padded LDS row stride (floats): 256 + 4 pad DWORDs
#define CENTER   127.5f       // (RES-1)/2
#define LDS_BYTES (RES * STRIDE * 4)

typedef __attribute__((ext_vector_type(2))) float v2f;
typedef __attribute__((ext_vector_type(8))) float v8f;
typedef __attribute__((ext_vector_type(4))) unsigned int u32x4;
typedef __attribute__((ext_vector_type(4))) int i32x4;
typedef __attribute__((ext_vector_type(8))) int i32x8;

__global__ __launch_bounds__(256)
void radon_wmma_kernel(const float* __restrict__ imgs,
                       const float* __restrict__ angles,
                       float* __restrict__ out)
{
    extern __shared__ float lds[];

    const int tid  = threadIdx.x;
    const int wg   = blockIdx.x;
    const int b    = wg >> 5;      // batch index        (8)
    const int ablk = wg & 31;      // angle block index  (32 blocks of 8)

    // ---- Stage image b into LDS via the Tensor Data Mover ----
    // One TDM descriptor moves the whole 256x256 f32 tile; hardware padding
    // (4 DWORDs every 256 DWORDs) produces the 260-float bank-stagger stride.
    if (tid < 32) {                       // single wave issues the DMA
        const unsigned lds_base = (unsigned)(uintptr_t)(void*)lds;  // LDS byte offset
        const uint64_t ga = (uint64_t)(uintptr_t)(imgs + (size_t)b * (RES * RES));

        u32x4 g0;
        g0[0] = 1u;                                   // count=1, user mode, no gather
        g0[1] = lds_base;                             // lds_addr (bytes)
        g0[2] = (unsigned)(ga & 0xFFFFFFFFu);         // global_addr[31:0]
        g0[3] = (unsigned)((ga >> 32) & 0x01FFFFFFu)  // global_addr[56:32]
              | (2u << 30);                           // type = 2 ("image")

        i32x8 g1;
        g1[0] = (int)((2u << 16)      // data_size = 4 bytes
                    | (1u << 20)      // pad_enable
                    | (7u << 22)      // pad_interval: every 256 DWORDs
                    | (3u << 25));    // pad_amount: 4 DWORDs
        g1[1] = (int)(256u << 16);    // tensor_dim0[15:0]=256 (atomic_barrier_addr=0)
        g1[2] = (int)(256u << 16);    // tensor_dim0[31:16]=0 ; tensor_dim1[15:0]=256
        g1[3] = (int)(256u << 16);    // tensor_dim1[31:16]=0 ; tile_dim0=256
        g1[4] = (int)256u;            // tile_dim1=256 ; tile_dim2=0
        g1[5] = 256;                  // tensor_dim0_stride[31:0] = 256 elements
        g1[6] = 0;                    // stride hi / tensor_dim1_stride lo
        g1[7] = 0;                    // tensor_dim1_stride hi

        const i32x4 gz4 = {0, 0, 0, 0};               // 2-D tensor: groups 2/3 unused
        const i32x8 gz8 = {0, 0, 0, 0, 0, 0, 0, 0};   // trailing group (zeroed)
        __builtin_amdgcn_tensor_load_to_lds(g0, g1, gz4, gz4, gz8, 0);
        __builtin_amdgcn_s_wait_tensorcnt(0);
    }
    __syncthreads();

    const int lane = tid & 31;   // wave32
    const int wave = tid >> 5;   // 8 waves
    const int half = lane >> 4;  // K-half of the A fragment
    const int m    = lane & 15;  // matrix row within the s-tile

    const v2f ones = {1.0f, 1.0f};   // B matrix (4x16 all ones) -> row-sum reduction

    for (int j = 0; j < A_BLK; ++j) {
        const int   a   = ablk * A_BLK + j;
        const float ang = angles[a];
        const float cv  = cosf(ang);
        const float sv  = sinf(ang);

        // each wave: two 16-row s-tiles -> 8 waves cover all 256 s
        for (int tile = 0; tile < 2; ++tile) {
            const int   s_base = wave * 16 + tile * 128;
            const float s_cen  = (float)(s_base + m) - CENTER;
            const float bx     = fmaf(s_cen, cv, CENTER);  // x + c at t_cen = 0
            const float by     = fmaf(s_cen, sv, CENTER);  // y + c at t_cen = 0

            v8f acc = {};  // D/C accumulator: per-row ray sums in every column

            for (int i = 0; i < 64; ++i) {
                // this lane supplies K = 2*half + {0,1}  ->  t index 4*i + 2*half + k
                const float t0 = (float)(4 * i + 2 * half) - CENTER;
                v2f afrag;
                #pragma unroll
                for (int k = 0; k < 2; ++k) {
                    const float t_cen = t0 + (float)k;
                    const float xf = fmaf(-t_cen, sv, bx);   // col coordinate
                    const float yf = fmaf( t_cen, cv, by);   // row coordinate
                    const float xfl = floorf(xf);
                    const float yfl = floorf(yf);
                    const float fx = xf - xfl;
                    const float fy = yf - yfl;
                    const int x0 = (int)xfl, y0 = (int)yfl;
                    const int x1 = x0 + 1,  y1 = y0 + 1;
                    // zero-border bilinear: kill weights outside, clamp addresses
                    const float wx0 = ((unsigned)x0 < (unsigned)RES) ? (1.0f - fx) : 0.0f;
                    const float wx1 = ((unsigned)x1 < (unsigned)RES) ? fx          : 0.0f;
                    const float wy0 = ((unsigned)y0 < (unsigned)RES) ? (1.0f - fy) : 0.0f;
                    const float wy1 = ((unsigned)y1 < (unsigned)RES) ? fy          : 0.0f;
                    const int cx0 = min(max(x0, 0), RES - 1);
                    const int cx1 = min(max(x1, 0), RES - 1);
                    const int cy0 = min(max(y0, 0), RES - 1);
                    const int cy1 = min(max(y1, 0), RES - 1);
                    const float p00 = lds[cy0 * STRIDE + cx0];
                    const float p01 = lds[cy0 * STRIDE + cx1];
                    const float p10 = lds[cy1 * STRIDE + cx0];
                    const float p11 = lds[cy1 * STRIDE + cx1];
                    afrag[k] = fmaf(wy1, fmaf(wx1, p11, wx0 * p10),
                                    wy0 * fmaf(wx1, p01, wx0 * p00));
                }
                // D(16x16) += A(16x4) * ones(4x16): column n = running ray sum of row m
                acc = __builtin_amdgcn_wmma_f32_16x16x4_f32(
                        false, afrag, false, ones, (short)0, acc, false, false);
            }

            // D layout: VGPR r, lanes 0-15 -> D[r, lane]; lanes 16-31 -> D[r+8, lane-16].
            // Column 0 of D lives in lane 0 (rows 0..7) and lane 16 (rows 8..15).
            if (m == 0) {
                float* orow = out + (((size_t)b * A_TOT + a) * RES) + s_base + half * 8;
                float4 lo = {acc[0], acc[1], acc[2], acc[3]};
                float4 hi = {acc[4], acc[5], acc[6], acc[7]};
                *(float4*)&orow[0] = lo;
                *(float4*)&orow[4] = hi;
            }
        }
    }
}

extern "C" void kernel_launch(void* const* d_in, const int* in_sizes, int n_in,
                              void* d_out, int out_size, void* d_ws, size_t ws_size,
                              hipStream_t stream) {
    const float* imgs   = (const float*)d_in[0];   // [8,256,256] f32
    const float* angles = (const float*)d_in[1];   // [256] f32
    float* out          = (float*)d_out;           // [8,256,256] f32

    (void)in_sizes; (void)n_in; (void)out_size; (void)d_ws; (void)ws_size;

    // 266,240 B dynamic LDS per workgroup (fits CDNA5's 320 KB/WGP)
    (void)hipFuncSetAttribute((const void*)radon_wmma_kernel,
                              hipFuncAttributeMaxDynamicSharedMemorySize, LDS_BYTES);

    radon_wmma_kernel<<<dim3(BATCH * 32), dim3(256), LDS_BYTES, stream>>>(imgs, angles, out);
}